// GAT_58488864637047
// MI455X (gfx1250) — compile-verified
//
#include <hip/hip_runtime.h>
#include <math.h>

#define NN 100000
#define FIN 500
#define HID1 64            // 8 heads * 8 dim
#define HEADS 8
#define DIM 8
#define COUT 3
#define NEG_SLOPE 0.2f
#define EPSV 1e-16f

typedef __attribute__((ext_vector_type(2))) float v2f;
typedef __attribute__((ext_vector_type(8))) float v8f;

static __host__ __device__ inline int cdiv(int a, int b) { return (a + b - 1) / b; }

// ---------------------------------------------------------------- utilities
__global__ __launch_bounds__(256) void gat_fill(float* __restrict__ p, float v, int n) {
    int i = blockIdx.x * blockDim.x + threadIdx.x;
    if (i < n) p[i] = v;
}

__device__ inline void atomicMaxFloat(float* addr, float value) {
    // sign-aware int trick: works for any mix of signs, init must be -inf
    if (value >= 0.0f)
        atomicMax((int*)addr, __float_as_int(value));
    else
        atomicMin((unsigned int*)addr, __float_as_uint(value));
}

// ------------------------------------------------- GEMM1: h1 = x @ W1 (WMMA)
// x: [N,500] row-major, W: [500,64] row-major, h1: [N,64]
// one wave computes a 16(M) x 64(N) strip via 4 accumulators
__global__ __launch_bounds__(256) void gat_gemm1_wmma(const float* __restrict__ x,
                                                      const float* __restrict__ W,
                                                      float* __restrict__ h1) {
    const int wave = threadIdx.x >> 5;
    const int tile = blockIdx.x * 8 + wave;       // M tile index
    if (tile * 16 >= NN) return;                  // wave-uniform: EXEC stays all-ones
    const int lane = threadIdx.x & 31;
    const int half = lane >> 4;                   // 0: K={0,1}, 1: K={2,3}
    const int l15  = lane & 15;

    const float* xrow = x + (size_t)(tile * 16 + l15) * FIN;

    v8f acc0 = {}, acc1 = {}, acc2 = {}, acc3 = {};
    for (int k = 0; k < FIN; k += 4) {
        v2f a;
        a.x = xrow[k + 2 * half];
        a.y = xrow[k + 2 * half + 1];
        const float* wk = W + (size_t)(k + 2 * half) * HID1 + l15;
        v2f b0, b1, b2, b3;
        b0.x = wk[0];   b0.y = wk[HID1];
        b1.x = wk[16];  b1.y = wk[HID1 + 16];
        b2.x = wk[32];  b2.y = wk[HID1 + 32];
        b3.x = wk[48];  b3.y = wk[HID1 + 48];
        acc0 = __builtin_amdgcn_wmma_f32_16x16x4_f32(false, a, false, b0, (short)0, acc0, false, false);
        acc1 = __builtin_amdgcn_wmma_f32_16x16x4_f32(false, a, false, b1, (short)0, acc1, false, false);
        acc2 = __builtin_amdgcn_wmma_f32_16x16x4_f32(false, a, false, b2, (short)0, acc2, false, false);
        acc3 = __builtin_amdgcn_wmma_f32_16x16x4_f32(false, a, false, b3, (short)0, acc3, false, false);
    }
    // D layout: VGPR r -> rows r (lanes 0-15) / r+8 (lanes 16-31), N = lane&15
    for (int r = 0; r < 8; ++r) {
        int row = tile * 16 + r + half * 8;
        float* o = h1 + (size_t)row * HID1 + l15;
        o[0]  = acc0[r];
        o[16] = acc1[r];
        o[32] = acc2[r];
        o[48] = acc3[r];
    }
}

// ---------------------------------------- layer1 per-node attention logits
__global__ __launch_bounds__(256) void gat_logits1(const float* __restrict__ h1,
                                                   const float* __restrict__ att_src,
                                                   const float* __restrict__ att_dst,
                                                   float* __restrict__ alS,
                                                   float* __restrict__ alD) {
    int t = blockIdx.x * blockDim.x + threadIdx.x;   // [0, N*HEADS)
    if (t >= NN * HEADS) return;
    int n = t >> 3, h = t & 7;
    const float* hp = h1 + (size_t)n * HID1 + h * DIM;
    float s = 0.f, d = 0.f;
#pragma unroll
    for (int j = 0; j < DIM; ++j) {
        float v = hp[j];
        s += v * att_src[h * DIM + j];
        d += v * att_dst[h * DIM + j];
    }
    alS[t] = s;
    alD[t] = d;
}

// ------------------------------------------------------- layer1 edge passes
__device__ inline void resolve_edge(int i, int E, const int* __restrict__ src,
                                    const int* __restrict__ dst, int& s, int& d) {
    if (i < E) { s = src[i]; d = dst[i]; }
    else       { s = i - E; d = i - E; }           // virtual self-loops
}

__global__ __launch_bounds__(256) void gat_edge_max1(const int* __restrict__ src,
                                                     const int* __restrict__ dst, int E, int Etot,
                                                     const float* __restrict__ alS,
                                                     const float* __restrict__ alD,
                                                     float* __restrict__ emax) {
    int i = blockIdx.x * blockDim.x + threadIdx.x;
    if (i >= Etot) return;
    int s, d; resolve_edge(i, E, src, dst, s, d);
    const float4* ps = (const float4*)(alS + (size_t)s * HEADS);
    const float4* pd = (const float4*)(alD + (size_t)d * HEADS);
    float a[8], b[8];
    *(float4*)(a) = ps[0]; *(float4*)(a + 4) = ps[1];
    *(float4*)(b) = pd[0]; *(float4*)(b + 4) = pd[1];
#pragma unroll
    for (int h = 0; h < HEADS; ++h) {
        float e = a[h] + b[h];
        e = (e >= 0.f) ? e : NEG_SLOPE * e;
        atomicMaxFloat(emax + (size_t)d * HEADS + h, e);
    }
}

__global__ __launch_bounds__(256) void gat_edge_sum1(const int* __restrict__ src,
                                                     const int* __restrict__ dst, int E, int Etot,
                                                     const float* __restrict__ alS,
                                                     const float* __restrict__ alD,
                                                     const float* __restrict__ emax,
                                                     float* __restrict__ denom) {
    int i = blockIdx.x * blockDim.x + threadIdx.x;
    if (i >= Etot) return;
    int s, d; resolve_edge(i, E, src, dst, s, d);
    const float4* ps = (const float4*)(alS + (size_t)s * HEADS);
    const float4* pd = (const float4*)(alD + (size_t)d * HEADS);
    const float4* pm = (const float4*)(emax + (size_t)d * HEADS);
    float a[8], b[8], m[8];
    *(float4*)(a) = ps[0]; *(float4*)(a + 4) = ps[1];
    *(float4*)(b) = pd[0]; *(float4*)(b + 4) = pd[1];
    *(float4*)(m) = pm[0]; *(float4*)(m + 4) = pm[1];
#pragma unroll
    for (int h = 0; h < HEADS; ++h) {
        float e = a[h] + b[h];
        e = (e >= 0.f) ? e : NEG_SLOPE * e;
        atomicAdd(denom + (size_t)d * HEADS + h, expf(e - m[h]));
    }
}

// one thread per (edge, head): 8 gathered floats of h1[src], 8 atomic adds
__global__ __launch_bounds__(256) void gat_edge_agg1(const int* __restrict__ src,
                                                     const int* __restrict__ dst, int E, int Etot,
                                                     const float* __restrict__ alS,
                                                     const float* __restrict__ alD,
                                                     const float* __restrict__ emax,
                                                     const float* __restrict__ denom,
                                                     const float* __restrict__ h1,
                                                     float* __restrict__ out1) {
    int t = blockIdx.x * blockDim.x + threadIdx.x;   // [0, Etot*HEADS)
    if (t >= Etot * HEADS) return;
    int i = t >> 3, h = t & 7;
    int s, d; resolve_edge(i, E, src, dst, s, d);
    float e = alS[(size_t)s * HEADS + h] + alD[(size_t)d * HEADS + h];
    e = (e >= 0.f) ? e : NEG_SLOPE * e;
    float ee = expf(e - emax[(size_t)d * HEADS + h]);
    float alpha = ee / (denom[(size_t)d * HEADS + h] + EPSV);
    const float4* hs = (const float4*)(h1 + (size_t)s * HID1 + h * DIM);
    float v[8];
    *(float4*)(v) = hs[0]; *(float4*)(v + 4) = hs[1];
    float* od = out1 + (size_t)d * HID1 + h * DIM;
#pragma unroll
    for (int j = 0; j < DIM; ++j) atomicAdd(od + j, v[j] * alpha);
}

// ------------------------------------------- layer1 finalize: +bias, ELU
__global__ __launch_bounds__(256) void gat_bias_elu(float* __restrict__ out1,
                                                    const float* __restrict__ b1) {
    int t = blockIdx.x * blockDim.x + threadIdx.x;
    if (t >= NN * HID1) return;
    float v = out1[t] + b1[t & (HID1 - 1)];
    out1[t] = (v > 0.f) ? v : expm1f(v);
}

// -------------------------- GEMM2 + layer2 logits (tiny: thread per node)
__global__ __launch_bounds__(256) void gat_gemm2(const float* __restrict__ hact,
                                                 const float* __restrict__ W2,
                                                 const float* __restrict__ att_src2,
                                                 const float* __restrict__ att_dst2,
                                                 float* __restrict__ h2,
                                                 float* __restrict__ alS2,
                                                 float* __restrict__ alD2) {
    int n = blockIdx.x * blockDim.x + threadIdx.x;
    if (n >= NN) return;
    const float* hp = hact + (size_t)n * HID1;
    float acc[COUT] = {0.f, 0.f, 0.f};
    for (int k = 0; k < HID1; ++k) {
        float v = hp[k];
#pragma unroll
        for (int c = 0; c < COUT; ++c) acc[c] += v * W2[k * COUT + c];
    }
    float s = 0.f, d = 0.f;
#pragma unroll
    for (int c = 0; c < COUT; ++c) {
        h2[(size_t)n * COUT + c] = acc[c];
        s += acc[c] * att_src2[c];
        d += acc[c] * att_dst2[c];
    }
    alS2[n] = s;
    alD2[n] = d;
}

// ------------------------------------------------------- layer2 edge passes
__global__ __launch_bounds__(256) void gat_edge_max2(const int* __restrict__ src,
                                                     const int* __restrict__ dst, int E, int Etot,
                                                     const float* __restrict__ alS,
                                                     const float* __restrict__ alD,
                                                     float* __restrict__ emax) {
    int i = blockIdx.x * blockDim.x + threadIdx.x;
    if (i >= Etot) return;
    int s, d; resolve_edge(i, E, src, dst, s, d);
    float e = alS[s] + alD[d];
    e = (e >= 0.f) ? e : NEG_SLOPE * e;
    atomicMaxFloat(emax + d, e);
}

__global__ __launch_bounds__(256) void gat_edge_sum2(const int* __restrict__ src,
                                                     const int* __restrict__ dst, int E, int Etot,
                                                     const float* __restrict__ alS,
                                                     const float* __restrict__ alD,
                                                     const float* __restrict__ emax,
                                                     float* __restrict__ denom) {
    int i = blockIdx.x * blockDim.x + threadIdx.x;
    if (i >= Etot) return;
    int s, d; resolve_edge(i, E, src, dst, s, d);
    float e = alS[s] + alD[d];
    e = (e >= 0.f) ? e : NEG_SLOPE * e;
    atomicAdd(denom + d, expf(e - emax[d]));
}

__global__ __launch_bounds__(256) void gat_edge_agg2(const int* __restrict__ src,
                                                     const int* __restrict__ dst, int E, int Etot,
                                                     const float* __restrict__ alS,
                                                     const float* __restrict__ alD,
                                                     const float* __restrict__ emax,
                                                     const float* __restrict__ denom,
                                                     const float* __restrict__ h2,
                                                     float* __restrict__ out2) {
    int i = blockIdx.x * blockDim.x + threadIdx.x;
    if (i >= Etot) return;
    int s, d; resolve_edge(i, E, src, dst, s, d);
    float e = alS[s] + alD[d];
    e = (e >= 0.f) ? e : NEG_SLOPE * e;
    float ee = expf(e - emax[d]);
    float alpha = ee / (denom[d] + EPSV);
#pragma unroll
    for (int c = 0; c < COUT; ++c)
        atomicAdd(out2 + (size_t)d * COUT + c, h2[(size_t)s * COUT + c] * alpha);
}

// ------------------------------ finalize: +b2, write out + log_softmax
__global__ __launch_bounds__(256) void gat_finalize(const float* __restrict__ out2,
                                                    const float* __restrict__ b2,
                                                    float* __restrict__ dout) {
    int n = blockIdx.x * blockDim.x + threadIdx.x;
    if (n >= NN) return;
    float o[COUT];
#pragma unroll
    for (int c = 0; c < COUT; ++c) o[c] = out2[(size_t)n * COUT + c] + b2[c];
    float m = fmaxf(o[0], fmaxf(o[1], o[2]));
    float z = expf(o[0] - m) + expf(o[1] - m) + expf(o[2] - m);
    float lse = logf(z) + m;
#pragma unroll
    for (int c = 0; c < COUT; ++c) {
        dout[(size_t)n * COUT + c] = o[c];
        dout[(size_t)NN * COUT + (size_t)n * COUT + c] = o[c] - lse;
    }
}

// =========================================================== host launcher
extern "C" void kernel_launch(void* const* d_in, const int* in_sizes, int n_in,
                              void* d_out, int out_size, void* d_ws, size_t ws_size,
                              hipStream_t stream) {
    const float* x        = (const float*)d_in[0];
    const int*   ei       = (const int*)d_in[1];     // [2, E] int32
    const float* W1       = (const float*)d_in[2];
    const float* att_src1 = (const float*)d_in[3];
    const float* att_dst1 = (const float*)d_in[4];
    const float* b1       = (const float*)d_in[5];
    const float* W2       = (const float*)d_in[6];
    const float* att_src2 = (const float*)d_in[7];
    const float* att_dst2 = (const float*)d_in[8];
    const float* b2       = (const float*)d_in[9];
    float* out = (float*)d_out;

    const int E    = in_sizes[1] / 2;
    const int Etot = E + NN;
    const int* src = ei;
    const int* dst = ei + E;

    // ---- workspace layout (fp32), ~68 MB
    float* ws     = (float*)d_ws;
    float* h1     = ws;                             // N*64
    float* alS1   = h1    + (size_t)NN * HID1;      // N*8
    float* alD1   = alS1  + (size_t)NN * HEADS;
    float* emax1  = alD1  + (size_t)NN * HEADS;
    float* denom1 = emax1 + (size_t)NN * HEADS;
    float* out1   = denom1 + (size_t)NN * HEADS;    // N*64
    float* h2     = out1  + (size_t)NN * HID1;      // N*3
    float* alS2   = h2    + (size_t)NN * COUT;      // N
    float* alD2   = alS2  + NN;
    float* emax2  = alD2  + NN;
    float* denom2 = emax2 + NN;
    float* out2   = denom2 + NN;                    // N*3

    const int B = 256;

    // ---- init accumulators
    gat_fill<<<cdiv(NN * HEADS, B), B, 0, stream>>>(emax1, -INFINITY, NN * HEADS);
    gat_fill<<<cdiv(NN * HEADS, B), B, 0, stream>>>(denom1, 0.f, NN * HEADS);
    gat_fill<<<cdiv(NN * HID1, B), B, 0, stream>>>(out1, 0.f, NN * HID1);
    gat_fill<<<cdiv(NN, B), B, 0, stream>>>(emax2, -INFINITY, NN);
    gat_fill<<<cdiv(NN, B), B, 0, stream>>>(denom2, 0.f, NN);
    gat_fill<<<cdiv(NN * COUT, B), B, 0, stream>>>(out2, 0.f, NN * COUT);

    // ---- layer 1
    gat_gemm1_wmma<<<cdiv(NN / 16, 8), 256, 0, stream>>>(x, W1, h1);
    gat_logits1<<<cdiv(NN * HEADS, B), B, 0, stream>>>(h1, att_src1, att_dst1, alS1, alD1);
    gat_edge_max1<<<cdiv(Etot, B), B, 0, stream>>>(src, dst, E, Etot, alS1, alD1, emax1);
    gat_edge_sum1<<<cdiv(Etot, B), B, 0, stream>>>(src, dst, E, Etot, alS1, alD1, emax1, denom1);
    gat_edge_agg1<<<cdiv(Etot * HEADS, B), B, 0, stream>>>(src, dst, E, Etot, alS1, alD1,
                                                           emax1, denom1, h1, out1);
    gat_bias_elu<<<cdiv(NN * HID1, B), B, 0, stream>>>(out1, b1);

    // ---- layer 2
    gat_gemm2<<<cdiv(NN, B), B, 0, stream>>>(out1, W2, att_src2, att_dst2, h2, alS2, alD2);
    gat_edge_max2<<<cdiv(Etot, B), B, 0, stream>>>(src, dst, E, Etot, alS2, alD2, emax2);
    gat_edge_sum2<<<cdiv(Etot, B), B, 0, stream>>>(src, dst, E, Etot, alS2, alD2, emax2, denom2);
    gat_edge_agg2<<<cdiv(Etot, B), B, 0, stream>>>(src, dst, E, Etot, alS2, alD2, emax2, denom2,
                                                   h2, out2);

    // ---- outputs: [out ; log_softmax(out)]
    gat_finalize<<<cdiv(NN, B), B, 0, stream>>>(out2, b2, out);
}